// DecoderOnlyAspire_85040352461391
// MI455X (gfx1250) — compile-verified
//
#include <hip/hip_runtime.h>
#include <hip/hip_bf16.h>

typedef __attribute__((ext_vector_type(2))) float v2f;
typedef __attribute__((ext_vector_type(8))) float v8f;

#define B_ 8
#define S_ 4096
#define D_ 1536
#define M_ 24
#define NTILES (D_ / 16)                 // 96 column tiles of 16
#define WAVES_PER_BLOCK 4
#define NT_GROUPS (NTILES / WAVES_PER_BLOCK)  // 24
#define SCHUNKS 8
#define SCHUNK (S_ / SCHUNKS)            // 512 tokens per wave

// ---------------------------------------------------------------- zero scratch
__global__ void zero_f32_kernel(float* __restrict__ p, int n) {
    int i = blockIdx.x * blockDim.x + threadIdx.x;
    if (i < n) p[i] = 0.0f;
}

// ---------------------------------------------------- counts / seq-len / last
__global__ void count_kernel(const int* __restrict__ ids,
                             const int* __restrict__ mask,
                             float* __restrict__ cnts,   // [B][M]
                             int*   __restrict__ seqlen, // [B]
                             int*   __restrict__ lastv)  // [B]
{
    int b = blockIdx.x;
    __shared__ int cnt[M_];
    __shared__ int msum;
    if (threadIdx.x < M_) cnt[threadIdx.x] = 0;
    if (threadIdx.x == 0) msum = 0;
    __syncthreads();

    int lm = 0;
    for (int s = threadIdx.x; s < S_; s += blockDim.x) {
        atomicAdd(&cnt[ids[b * S_ + s]], 1);   // ds_add_u32
        lm += mask[b * S_ + s];
    }
    atomicAdd(&msum, lm);
    __syncthreads();

    if (threadIdx.x < M_) cnts[b * M_ + threadIdx.x] = (float)cnt[threadIdx.x];
    if (threadIdx.x == 0) {
        seqlen[b] = msum;
        lastv[b]  = mask[b * S_ + (S_ - 1)];
    }
}

// ------------------------------------------- segmented sum as one-hot x hidden
// D(16x16) = A(one-hot 16x4) x B(hidden 4x16) + C, accumulated over a 512-token
// S-chunk. Two m-tiles (rows 0-15, 16-31) share each loaded B fragment so
// `hidden` is streamed exactly once.
__global__ void __launch_bounds__(128)
segsum_wmma_kernel(const float* __restrict__ hidden,  // [B][S][D]
                   const int*   __restrict__ ids,     // [B][S]
                   float*       __restrict__ sums)    // [B][M_][D]
{
    const int lane   = threadIdx.x & 31;
    const int wave   = threadIdx.x >> 5;
    const int ntile  = blockIdx.x * WAVES_PER_BLOCK + wave;   // 0..95
    const int sbase  = blockIdx.y * SCHUNK;
    const int b      = blockIdx.z;

    const int lane16 = lane & 15;   // A: row m within tile; B: column n; C: col n
    const int kh     = lane >> 4;   // which K-half this lane holds

    const float* __restrict__ hcol = hidden + (size_t)b * S_ * D_ + ntile * 16 + lane16;
    const int*   __restrict__ irow = ids + b * S_;

    const int m0 = lane16;          // one-hot row for m-tile 0 (m = 0..15)
    const int m1 = 16 + lane16;     // one-hot row for m-tile 1 (m = 16..31)

    v8f c0 = {};
    v8f c1 = {};

    for (int s = sbase; s < sbase + SCHUNK; s += 4) {
        const int s0 = s + 2 * kh;          // this lane's two K slots
        const int i0 = irow[s0];
        const int i1 = irow[s0 + 1];

        v2f a0, a1, bb;
        a0.x = (i0 == m0) ? 1.0f : 0.0f;
        a0.y = (i1 == m0) ? 1.0f : 0.0f;
        a1.x = (i0 == m1) ? 1.0f : 0.0f;
        a1.y = (i1 == m1) ? 1.0f : 0.0f;
        bb.x = hcol[(size_t)s0 * D_];
        bb.y = hcol[(size_t)(s0 + 1) * D_];

        c0 = __builtin_amdgcn_wmma_f32_16x16x4_f32(false, a0, false, bb,
                                                   (short)0, c0, false, false);
        c1 = __builtin_amdgcn_wmma_f32_16x16x4_f32(false, a1, false, bb,
                                                   (short)0, c1, false, false);
    }

    // Flush: C layout is m = r + 8*kh (+16 for tile 1), n = lane16.
    float* outb = sums + ((size_t)b * M_) * D_ + ntile * 16 + lane16;
#pragma unroll
    for (int r = 0; r < 8; ++r) {
        int m = r + 8 * kh;                 // 0..15
        atomicAdd(outb + (size_t)m * D_, c0[r]);
    }
    if (kh == 0) {                          // tile-1 rows 24..31 are exact zeros
#pragma unroll
        for (int r = 0; r < 8; ++r) {
            int m = 16 + r;                 // 16..23
            atomicAdd(outb + (size_t)m * D_, c1[r]);
        }
    }
}

// ------------------------------------------------------------- sent_reps = s/c
__global__ void finalize_sent_kernel(const float* __restrict__ sums,
                                     const float* __restrict__ cnts,
                                     float* __restrict__ out_sent) // [B][M][D]
{
    int bm = blockIdx.x;                    // 0..B*M-1
    float c = cnts[bm];
    float dv = fmaxf(c, 1.0f);
    for (int d = threadIdx.x; d < D_; d += blockDim.x)
        out_sent[(size_t)bm * D_ + d] = sums[(size_t)bm * D_ + d] / dv;
}

// ----------------------------------------------------------- last-token pool
__global__ void doc_kernel(const float* __restrict__ hidden,
                           const int* __restrict__ seqlen,
                           const int* __restrict__ lastv,
                           float* __restrict__ out_doc) // [B][D]
{
    int b = blockIdx.x;
    int lastsum = 0;
#pragma unroll
    for (int i = 0; i < B_; ++i) lastsum += lastv[i];
    int idx = (lastsum == B_) ? (S_ - 1) : (seqlen[b] - 1);
    const float* src = hidden + ((size_t)b * S_ + (size_t)idx) * D_;
    for (int d = threadIdx.x; d < D_; d += blockDim.x)
        out_doc[b * D_ + d] = src[d];
}

// ---------------------------------------------------------------------- launch
extern "C" void kernel_launch(void* const* d_in, const int* in_sizes, int n_in,
                              void* d_out, int out_size, void* d_ws, size_t ws_size,
                              hipStream_t stream) {
    const float* hidden = (const float*)d_in[0];   // [B,S,D] f32
    const int*   mask   = (const int*)d_in[1];     // [B,S] i32
    const int*   ids    = (const int*)d_in[2];     // [B,S] i32
    // d_in[3] = max_sents (scalar 24), baked in at compile time.

    float* out      = (float*)d_out;
    float* out_doc  = out;                 // [B,D]
    float* out_sent = out + B_ * D_;       // [B,M,D]

    char*  ws     = (char*)d_ws;
    float* sums   = (float*)ws;                                        // B*M*D
    float* cnts   = (float*)(ws + (size_t)B_ * M_ * D_ * sizeof(float)); // B*M
    int*   seqlen = (int*)((char*)cnts + B_ * M_ * sizeof(float));     // B
    int*   lastv  = seqlen + B_;                                       // B

    const int nz = B_ * M_ * D_;
    zero_f32_kernel<<<(nz + 255) / 256, 256, 0, stream>>>(sums, nz);

    count_kernel<<<B_, 256, 0, stream>>>(ids, mask, cnts, seqlen, lastv);

    dim3 grid(NT_GROUPS, SCHUNKS, B_);     // 24 x 8 x 8 blocks, 4 waves each
    segsum_wmma_kernel<<<grid, 128, 0, stream>>>(hidden, ids, sums);

    finalize_sent_kernel<<<B_ * M_, 256, 0, stream>>>(sums, cnts, out_sent);
    doc_kernel<<<B_, 256, 0, stream>>>(hidden, seqlen, lastv, out_doc);
}